// TripletLoss_29721173688752
// MI455X (gfx1250) — compile-verified
//
#include <hip/hip_runtime.h>

typedef __attribute__((ext_vector_type(2))) float v2f;
typedef __attribute__((ext_vector_type(8))) float v8f;

#define NN 4096
#define DD 128
#define KRET 7
#define MARGIN_F 0.2f

// ---------------- Kernel 0: diag[i] = dot(A_i, B_i); zero accumulators ----------------
__global__ void diag_init_kernel(const float* __restrict__ A,
                                 const float* __restrict__ B,
                                 float* __restrict__ ws) {
  int i = blockIdx.x * blockDim.x + threadIdx.x;
  if (i < NN) {
    const float4* a = (const float4*)(A + (size_t)i * DD);
    const float4* b = (const float4*)(B + (size_t)i * DD);
    float s = 0.f;
#pragma unroll
    for (int k = 0; k < DD / 4; ++k) {
      float4 av = a[k], bv = b[k];
      s += av.x * bv.x + av.y * bv.y + av.z * bv.z + av.w * bv.w;
    }
    ws[16 + i] = s;
  }
  if (blockIdx.x == 0 && threadIdx.x == 0) { ws[0] = 0.f; ws[1] = 0.f; }
}

// ---------------- Kernel 1: fused WMMA distance strip + top-k + loss ----------------
__launch_bounds__(256, 1)
__global__ void triplet_main_kernel(const float* __restrict__ A,
                                    const float* __restrict__ B1,
                                    float* __restrict__ ws) {
  extern __shared__ float dist[];            // 16 rows x 4096 cols = 256 KB LDS
  const float* diagG = ws + 16;
  const int lane  = threadIdx.x & 31;
  const int wave  = threadIdx.x >> 5;
  const int r0    = blockIdx.x << 4;         // 16 rows per workgroup
  const int rowA  = r0 + (lane & 15);
  const int klo   = (lane < 16) ? 0 : 2;     // 16x16x4 f32 A-frag K placement
  const int half8 = (lane < 16) ? 0 : 8;     // C-frag M placement

  // ---- Phase 1: C = A(16xK) x B1^T tiles via V_WMMA_F32_16X16X4_F32 ----
  v2f afrag[32];
  const float* arow = A + (size_t)rowA * DD;
#pragma unroll
  for (int kk = 0; kk < 32; ++kk)
    afrag[kk] = *(const v2f*)(arow + kk * 4 + klo);

  float dI[8];
#pragma unroll
  for (int v = 0; v < 8; ++v) dI[v] = diagG[r0 + half8 + v];

  // B fragment ping-pong staging: burst-load 32 x b64 per tile, overlap with WMMA chain
  v2f bf0[32], bf1[32];
  const float* bbase = B1 + (size_t)((wave << 9) + (lane & 15)) * DD + klo;

#define LOADB(t, buf)                                                     \
  {                                                                       \
    const float* _p = bbase + (size_t)(t) * 16 * DD;                      \
    _Pragma("unroll")                                                     \
    for (int kk = 0; kk < 32; ++kk) (buf)[kk] = *(const v2f*)(_p + kk * 4); \
  }

#define DOTILE(t, buf)                                                    \
  {                                                                       \
    v8f acc = {};                                                         \
    _Pragma("unroll")                                                     \
    for (int kk = 0; kk < 32; ++kk)                                       \
      acc = __builtin_amdgcn_wmma_f32_16x16x4_f32(false, afrag[kk], false,\
                (buf)[kk], (short)0, acc, false, false);                  \
    const int _col = (wave << 9) + ((t) << 4) + (lane & 15);              \
    const float _dJ = diagG[_col];                                        \
    _Pragma("unroll")                                                     \
    for (int v = 0; v < 8; ++v) {                                         \
      float _d2 = dI[v] + _dJ - 2.0f * acc[v];                            \
      dist[(half8 + v) * NN + _col] = sqrtf(fmaxf(_d2, 1e-4f));           \
    }                                                                     \
  }

  LOADB(0, bf0);
#pragma unroll 1
  for (int t = 0; t < 32; t += 2) {
    LOADB(t + 1, bf1);                 // tile t+1 loads in flight...
    DOTILE(t, bf0);                    // ...while tile t WMMA chain runs
    if (t + 2 < 32) LOADB(t + 2, bf0); // tile t+2 loads in flight...
    DOTILE(t + 1, bf1);                // ...while tile t+1 WMMA chain runs
  }
#undef LOADB
#undef DOTILE
  __syncthreads();

  // ---- Phase 2: per-row top-7 negatives + triplet loss (2 rows per wave) ----
  float lsum = 0.f, lcnt = 0.f;
  for (int rr = 0; rr < 2; ++rr) {
    const int m   = wave + (rr << 3);
    const int i   = r0 + m;
    const int grp = i >> 3;
    const float* drow = dist + (size_t)m * NN;

    // per-lane 7 smallest (val,idx) over strided slice, skipping positives
    float bv[KRET]; int bi[KRET];
#pragma unroll
    for (int k = 0; k < KRET; ++k) { bv[k] = 3.0e38f; bi[k] = 0x7FFFFFFF; }
    for (int j = lane; j < NN; j += 32) {
      if ((j >> 3) == grp && j != i) continue;       // positive: masked out
      float v = drow[j];
      int w = 0;
#pragma unroll
      for (int k = 1; k < KRET; ++k)
        if (bv[k] > bv[w] || (bv[k] == bv[w] && bi[k] > bi[w])) w = k;
      if (v < bv[w] || (v == bv[w] && j < bi[w])) { bv[w] = v; bi[w] = j; }
    }
    // local ascending sort by (val, idx)
#pragma unroll
    for (int a = 1; a < KRET; ++a) {
      float v = bv[a]; int id = bi[a]; int b = a - 1;
      while (b >= 0 && (bv[b] > v || (bv[b] == v && bi[b] > id))) {
        bv[b + 1] = bv[b]; bi[b + 1] = bi[b]; --b;
      }
      bv[b + 1] = v; bi[b + 1] = id;
    }
    // wave-wide 7-round argmin merge (lowest-index tie-break == top_k semantics)
    int head = 0;
    int sel[KRET];
    for (int k = 0; k < KRET; ++k) {
      float mv = 3.0e38f; int mi = 0x7FFFFFFF;
      if (head < KRET) { mv = bv[head]; mi = bi[head]; }
      for (int off = 16; off > 0; off >>= 1) {
        float ov = __shfl_xor(mv, off, 32);
        int   oi = __shfl_xor(mi, off, 32);
        if (ov < mv || (ov == mv && oi < mi)) { mv = ov; mi = oi; }
      }
      sel[k] = mi;
      if (head < KRET && bi[head] == mi) ++head;     // owner lane pops
    }
    // negatives index-sorted ascending (neg_sorted)
#pragma unroll
    for (int a = 1; a < KRET; ++a) {
      int id = sel[a]; int b = a - 1;
      while (b >= 0 && sel[b] > id) { sel[b + 1] = sel[b]; --b; }
      sel[b + 1] = id;
    }
    // loss: d_ap/d_an from embeddings A only (per reference), 4 dims/lane
    const float4 ai = ((const float4*)(A + (size_t)i * DD))[lane];
    const int r = i & 7;
    for (int k = 0; k < KRET; ++k) {
      const int pk = (grp << 3) + (k < r ? k : k + 1);   // k-th positive (ascending)
      const int nk = sel[k];
      const float dpos = drow[pk];
      const float dneg = drow[nk];
      const float4 ap = ((const float4*)(A + (size_t)pk * DD))[lane];
      const float4 an = ((const float4*)(A + (size_t)nk * DD))[lane];
      float dx = ap.x - ai.x, dy = ap.y - ai.y, dz = ap.z - ai.z, dw = ap.w - ai.w;
      float ps = dx * dx + dy * dy + dz * dz + dw * dw;
      dx = an.x - ai.x; dy = an.y - ai.y; dz = an.z - ai.z; dw = an.w - ai.w;
      float ns = dx * dx + dy * dy + dz * dz + dw * dw;
      for (int off = 16; off > 0; off >>= 1) {
        ps += __shfl_xor(ps, off, 32);
        ns += __shfl_xor(ns, off, 32);
      }
      if (dpos < dneg + MARGIN_F) {                     // valid (has_pos always true)
        float d_ap = sqrtf(ps + 1e-8f);
        float d_an = sqrtf(ns + 1e-8f);
        lsum += (d_ap + MARGIN_F) + fmaxf(MARGIN_F - d_an, 0.f);
        lcnt += 1.f;
      }
    }
  }
  if (lane == 0) {
    atomicAdd(&ws[0], lsum);
    atomicAdd(&ws[1], lcnt);
  }
}

// ---------------- Kernel 2: finalize scalar ----------------
__global__ void finalize_kernel(const float* __restrict__ ws, float* __restrict__ out) {
  float ls = ws[0], c = ws[1];
  out[0] = (c > 0.f) ? (ls / c) : ls;
}

extern "C" void kernel_launch(void* const* d_in, const int* in_sizes, int n_in,
                              void* d_out, int out_size, void* d_ws, size_t ws_size,
                              hipStream_t stream) {
  const float* A  = (const float*)d_in[0];   // embeddings
  const float* B1 = (const float*)d_in[2];   // embeddings1
  float* ws  = (float*)d_ws;
  float* out = (float*)d_out;

  diag_init_kernel<<<(NN + 255) / 256, 256, 0, stream>>>(A, B1, ws);
  const size_t smem = (size_t)16 * NN * sizeof(float);   // 256 KB LDS strip
  triplet_main_kernel<<<NN / 16, 256, smem, stream>>>(A, B1, ws);
  finalize_kernel<<<1, 1, 0, stream>>>(ws, out);
}